// GATv2_40132174414163
// MI455X (gfx1250) — compile-verified
//
#include <hip/hip_runtime.h>
#include <hip/hip_bf16.h>

// ---------------------------------------------------------------------------
// GATv2 link-predictor forward on gfx1250 (wave32, WMMA f32 16x16x4).
// ---------------------------------------------------------------------------

typedef __attribute__((ext_vector_type(2))) float v2f;
typedef __attribute__((ext_vector_type(8))) float v8f;

#define NN_NODES 25000
#define N_EDGES  400000
#define N_PAIR   100000
#define N_ROWS   200000   // pos + neg pairs stacked

// ---- order-preserving float<->uint encoding for atomic max -----------------
__device__ __forceinline__ unsigned enc_f32(float f) {
  unsigned u = __float_as_uint(f);
  return (u & 0x80000000u) ? ~u : (u | 0x80000000u);
}
__device__ __forceinline__ float dec_f32(unsigned e) {
  return (e & 0x80000000u) ? __uint_as_float(e & 0x7FFFFFFFu)
                           : __uint_as_float(~e);
}
// enc(-inf) = ~0xFF800000 = 0x007FFFFF
#define ENC_NEG_INF 0x007FFFFFu

__device__ __forceinline__ float leaky(float v, float slope) {
  return v >= 0.f ? v : v * slope;
}

// ---------------------------------------------------------------------------
// Generic f32 GEMM:  C[M,N] = A[M,K] @ W[N,K]^T (+bias) (+leaky)
// Block: 256 threads = 8 waves; tile 64(M) x 32(N); BLK_K = 32.
// Wave (mw,nw) owns one 16x16 tile -> V_WMMA_F32_16X16X4_F32, f32 accumulate.
// Staging is float4 (global_load_b128 -> ds_store_b128), branch-free via
// index clamping + select. LDS pitch 36 floats: 16B aligned, bank-cycle 16.
// Requires: N % 32 == 0, K % 32 == 0. M ragged (guarded by clamp/select).
// ---------------------------------------------------------------------------
__global__ void gemm_wmma_f32(const float* __restrict__ A,
                              const float* __restrict__ W,
                              const float* __restrict__ bias,
                              float* __restrict__ C,
                              int M, int N, int K,
                              float slope, int act) {
  __shared__ float As[64][36];
  __shared__ float Ws[32][36];

  const int t     = threadIdx.x;
  const int lane  = t & 31;
  const int wave  = t >> 5;
  const int waveM = wave & 3;       // 0..3  -> 16-row stripes
  const int waveN = wave >> 2;      // 0..1  -> 16-col stripes
  const int mBase = blockIdx.x * 64;
  const int nBase = blockIdx.y * 32;
  const int lh    = lane >> 4;      // half-wave select
  const int l15   = lane & 15;

  v8f acc = {0.f, 0.f, 0.f, 0.f, 0.f, 0.f, 0.f, 0.f};

  for (int k0 = 0; k0 < K; k0 += 32) {
    // ---- stage A: 64 rows x 32 cols = 512 float4, 2 per thread ----
#pragma unroll
    for (int i = 0; i < 2; ++i) {
      int idx = t + i * 256;              // 0..511
      int r   = idx >> 3;                 // row in tile
      int c4  = (idx & 7) * 4;            // float4 column
      int gr  = mBase + r;
      int grc = gr < M ? gr : M - 1;      // clamp -> unconditional load
      float4 v = *(const float4*)(A + (size_t)grc * K + k0 + c4);
      if (gr >= M) { v.x = 0.f; v.y = 0.f; v.z = 0.f; v.w = 0.f; }
      *(float4*)&As[r][c4] = v;
    }
    // ---- stage W: 32 rows x 32 cols = 256 float4, 1 per thread ----
    {
      int r  = t >> 3;
      int c4 = (t & 7) * 4;
      *(float4*)&Ws[r][c4] =
          *(const float4*)(W + (size_t)(nBase + r) * K + k0 + c4);
    }
    __syncthreads();

    // A 16x4 layout: lanes 0-15 K={0,1}, lanes 16-31 K={2,3} (VGPR0/1);
    // B  4x16 mirrored: n = lane&15, same K split.
#pragma unroll
    for (int ks = 0; ks < 8; ++ks) {
      int kb = ks * 4 + 2 * lh;
      v2f a, b;
      a.x = As[waveM * 16 + l15][kb];
      a.y = As[waveM * 16 + l15][kb + 1];
      b.x = Ws[waveN * 16 + l15][kb];
      b.y = Ws[waveN * 16 + l15][kb + 1];
      acc = __builtin_amdgcn_wmma_f32_16x16x4_f32(
          /*neg_a=*/false, a, /*neg_b=*/false, b,
          /*c_mod=*/(short)0, acc, /*reuse_a=*/false, /*reuse_b=*/false);
    }
    __syncthreads();
  }

  // C/D layout: lanes 0-15 hold M=r, lanes 16-31 hold M=r+8; N = lane&15.
#pragma unroll
  for (int r = 0; r < 8; ++r) {
    int row = mBase + waveM * 16 + r + 8 * lh;
    int col = nBase + waveN * 16 + l15;
    if (row < M) {
      float v = acc[r];
      if (bias) v += bias[col];
      if (act)  v = leaky(v, slope);
      C[(size_t)row * N + col] = v;
    }
  }
}

// ---------------------------------------------------------------------------
// p[n,h] = dot(ht[n, h*F : h*F+F], a)   (attention projection, linear split)
// ---------------------------------------------------------------------------
template <int H, int F>
__global__ void node_p_kernel(const float* __restrict__ ht,
                              const float* __restrict__ a,
                              float* __restrict__ p) {
  int idx = blockIdx.x * blockDim.x + threadIdx.x;
  if (idx >= NN_NODES * H) return;
  int n = idx / H, h = idx % H;
  const float* row = ht + ((size_t)n * H + h) * F;
  float acc = 0.f;
#pragma unroll 4
  for (int f = 0; f < F; f += 4) {
    float4 v = *(const float4*)(row + f);
    float4 w = *(const float4*)(a + f);
    acc += v.x * w.x + v.y * w.y + v.z * w.z + v.w * w.w;
  }
  p[idx] = acc;
}

__global__ void fill_u32_kernel(unsigned* __restrict__ dst, unsigned val, int n) {
  int i = blockIdx.x * blockDim.x + threadIdx.x;
  if (i < n) dst[i] = val;
}

// ---------------------------------------------------------------------------
// Per-edge score e = leaky(p[src]+p[dst]+log1p(ew), 0.01); atomic seg-max.
// ---------------------------------------------------------------------------
template <int H>
__global__ void edge_score_kernel(const float* __restrict__ p,
                                  const float* __restrict__ ew,
                                  const int* __restrict__ src,
                                  const int* __restrict__ dst,
                                  float* __restrict__ ebuf,
                                  unsigned* __restrict__ menc) {
  int e = blockIdx.x * blockDim.x + threadIdx.x;
  if (e >= N_EDGES) return;
  float lw = log1pf(ew[e]);
  int s = src[e], d = dst[e];
  float ps[H], pd[H];
#pragma unroll
  for (int h = 0; h < H; ++h) ps[h] = p[s * H + h];   // coalesces to b128 (H=4)
#pragma unroll
  for (int h = 0; h < H; ++h) pd[h] = p[d * H + h];
#pragma unroll
  for (int h = 0; h < H; ++h) {
    float sc = leaky(ps[h] + pd[h] + lw, 0.01f);
    ebuf[(size_t)e * H + h] = sc;
    atomicMax(&menc[d * H + h], enc_f32(sc));
  }
}

// decode seg-max in place; empty segments (-inf) -> 0 (isfinite guard)
__global__ void node_m_kernel(unsigned* __restrict__ menc, int n) {
  int i = blockIdx.x * blockDim.x + threadIdx.x;
  if (i >= n) return;
  float m = dec_f32(menc[i]);
  if (!isfinite(m)) m = 0.f;
  menc[i] = __float_as_uint(m);
}

// ex = exp(e - m[dst]); atomic seg-sum into s
template <int H>
__global__ void edge_exp_kernel(float* __restrict__ ebuf,
                                const float* __restrict__ mf,
                                const int* __restrict__ dst,
                                float* __restrict__ sden) {
  int idx = blockIdx.x * blockDim.x + threadIdx.x;
  if (idx >= N_EDGES * H) return;
  int e = idx / H, h = idx % H;                      // shifts (H pow2)
  int d = dst[e];
  float ex = __expf(ebuf[idx] - mf[d * H + h]);
  ebuf[idx] = ex;
  atomicAdd(&sden[d * H + h], ex);
}

// out[dst,h,:] += (ex / max(s[dst,h],1e-20)) * ht[src,h,:]
template <int H, int F>
__global__ void edge_agg_kernel(const float* __restrict__ ebuf,
                                const float* __restrict__ sden,
                                const float* __restrict__ ht,
                                const int* __restrict__ src,
                                const int* __restrict__ dst,
                                float* __restrict__ hagg) {
  int idx = blockIdx.x * blockDim.x + threadIdx.x;
  if (idx >= N_EDGES * H) return;
  int e = idx / H, h = idx % H;
  int s = src[e], d = dst[e];
  float alpha = ebuf[idx] / fmaxf(sden[d * H + h], 1e-20f);
  const float* hrow = ht + ((size_t)s * H + h) * F;
  float* orow = hagg + ((size_t)d * H + h) * F;
#pragma unroll 4
  for (int f = 0; f < F; f += 4) {
    float4 v = *(const float4*)(hrow + f);
    atomicAdd(orow + f + 0, alpha * v.x);
    atomicAdd(orow + f + 1, alpha * v.y);
    atomicAdd(orow + f + 2, alpha * v.z);
    atomicAdd(orow + f + 3, alpha * v.w);
  }
}

// out = (in + b[col&mask]) , optional leaky
__global__ void bias_act_kernel(const float* __restrict__ in,
                                const float* __restrict__ b,
                                float* __restrict__ out,
                                int total, int mask, float slope, int act) {
  int i = blockIdx.x * blockDim.x + threadIdx.x;
  if (i >= total) return;
  float v = in[i] + b[i & mask];
  if (act) v = leaky(v, slope);
  out[i] = v;
}

// Z[row, :] = emb[srcIdx]*emb[dstIdx], rows 0..99999 pos, 100000..199999 neg
__global__ void build_z_kernel(const float* __restrict__ emb,
                               const int* __restrict__ ps, const int* __restrict__ pd,
                               const int* __restrict__ ns, const int* __restrict__ nd,
                               float* __restrict__ Z) {
  int idx = blockIdx.x * blockDim.x + threadIdx.x;   // rows * (128/4)
  if (idx >= N_ROWS * 32) return;
  int row = idx >> 5;
  int f   = (idx & 31) * 4;
  int s, d;
  if (row < N_PAIR) { s = ps[row]; d = pd[row]; }
  else              { s = ns[row - N_PAIR]; d = nd[row - N_PAIR]; }
  float4 a = *(const float4*)(emb + (size_t)s * 128 + f);
  float4 b = *(const float4*)(emb + (size_t)d * 128 + f);
  float4 r; r.x = a.x * b.x; r.y = a.y * b.y; r.z = a.z * b.z; r.w = a.w * b.w;
  *(float4*)(Z + (size_t)row * 128 + f) = r;
}

// final 32 -> 1 projection
__global__ void final_dot_kernel(const float* __restrict__ Z2,
                                 const float* __restrict__ pW3,
                                 const float* __restrict__ pb3,
                                 float* __restrict__ out) {
  int row = blockIdx.x * blockDim.x + threadIdx.x;
  if (row >= N_ROWS) return;
  const float* z = Z2 + (size_t)row * 32;
  float acc = pb3[0];
#pragma unroll
  for (int j = 0; j < 32; j += 4) {
    float4 a = *(const float4*)(z + j);
    float4 w = *(const float4*)(pW3 + j);
    acc += a.x * w.x + a.y * w.y + a.z * w.z + a.w * w.w;
  }
  out[row] = acc;
}

// ---------------------------------------------------------------------------
extern "C" void kernel_launch(void* const* d_in, const int* in_sizes, int n_in,
                              void* d_out, int out_size, void* d_ws, size_t ws_size,
                              hipStream_t stream) {
  const float* x    = (const float*)d_in[0];
  const float* ew   = (const float*)d_in[1];
  const int*   esrc = (const int*)d_in[2];
  const int*   edst = (const int*)d_in[3];
  const int*   psrc = (const int*)d_in[4];
  const int*   pdst = (const int*)d_in[5];
  const int*   nsrc = (const int*)d_in[6];
  const int*   ndst = (const int*)d_in[7];
  const float* Wl[3] = {(const float*)d_in[8],  (const float*)d_in[11], (const float*)d_in[14]};
  const float* al[3] = {(const float*)d_in[9],  (const float*)d_in[12], (const float*)d_in[15]};
  const float* bl[3] = {(const float*)d_in[10], (const float*)d_in[13], (const float*)d_in[16]};
  const float* pW1 = (const float*)d_in[17]; const float* pb1 = (const float*)d_in[18];
  const float* pW2 = (const float*)d_in[19]; const float* pb2 = (const float*)d_in[20];
  const float* pW3 = (const float*)d_in[21]; const float* pb3 = (const float*)d_in[22];

  float* out   = (float*)d_out;
  float* emb   = out + 2 * N_PAIR;           // gat_embedding region [25000 x 128]

  // workspace layout (floats)
  float*    ws   = (float*)d_ws;
  float*    ht   = ws;                        // 25000*256
  float*    hbuf = ht   + 6400000;            // 25000*256
  float*    pbuf = hbuf + 6400000;            // 25000*4
  unsigned* menc = (unsigned*)(pbuf + 100000);// 25000*4 (max, reused as float)
  float*    sden = (float*)(menc + 100000);   // 25000*4
  float*    ebuf = sden + 100000;             // 400000*4
  float*    Z    = ebuf + 1600000;            // 200000*128
  float*    Z1   = Z    + 25600000;           // 200000*64
  float*    Z2   = Z1   + 12800000;           // 200000*32

  const int Hs[3]  = {4, 4, 1};
  const int Fs[3]  = {64, 64, 128};
  const int Kin[3] = {128, 256, 256};

  const float* hin = x;
  for (int l = 0; l < 3; ++l) {
    const int H = Hs[l], F = Fs[l], HF = H * F, K = Kin[l];
    float* hout = (l == 2) ? emb : hbuf;

    // h = hin @ W^T   (WMMA)
    dim3 gg((NN_NODES + 63) / 64, HF / 32);
    gemm_wmma_f32<<<gg, 256, 0, stream>>>(hin, Wl[l], nullptr, ht,
                                          NN_NODES, HF, K, 0.f, 0);
    // p[n,h] = <h[n,h,:], a>;  segment softmax init
    int nh = NN_NODES * H;
    if (l < 2) node_p_kernel<4, 64><<<(nh + 255) / 256, 256, 0, stream>>>(ht, al[l], pbuf);
    else       node_p_kernel<1, 128><<<(nh + 255) / 256, 256, 0, stream>>>(ht, al[l], pbuf);
    fill_u32_kernel<<<(nh + 255) / 256, 256, 0, stream>>>(menc, ENC_NEG_INF, nh);
    hipMemsetAsync(sden, 0, nh * sizeof(float), stream);

    if (l < 2) {
      edge_score_kernel<4><<<(N_EDGES + 255) / 256, 256, 0, stream>>>(
          pbuf, ew, esrc, edst, ebuf, menc);
    } else {
      edge_score_kernel<1><<<(N_EDGES + 255) / 256, 256, 0, stream>>>(
          pbuf, ew, esrc, edst, ebuf, menc);
    }
    node_m_kernel<<<(nh + 255) / 256, 256, 0, stream>>>(menc, nh);
    if (l < 2) {
      edge_exp_kernel<4><<<(N_EDGES * 4 + 255) / 256, 256, 0, stream>>>(
          ebuf, (const float*)menc, edst, sden);
    } else {
      edge_exp_kernel<1><<<(N_EDGES + 255) / 256, 256, 0, stream>>>(
          ebuf, (const float*)menc, edst, sden);
    }

    // weighted aggregation scatter
    hipMemsetAsync(hout, 0, (size_t)NN_NODES * HF * sizeof(float), stream);
    if (l < 2) {
      edge_agg_kernel<4, 64><<<(N_EDGES * 4 + 255) / 256, 256, 0, stream>>>(
          ebuf, sden, ht, esrc, edst, hout);
    } else {
      edge_agg_kernel<1, 128><<<(N_EDGES + 255) / 256, 256, 0, stream>>>(
          ebuf, sden, ht, esrc, edst, hout);
    }
    // + bias, leaky(0.01) between layers (not after the last)
    bias_act_kernel<<<(NN_NODES * HF + 255) / 256, 256, 0, stream>>>(
        hout, bl[l], hout, NN_NODES * HF, HF - 1, 0.01f, (l < 2) ? 1 : 0);
    hin = hout;
  }

  // predictor on pos/neg pairs: z = emb[s]*emb[d]; 128->64->32->1
  build_z_kernel<<<(N_ROWS * 32 + 255) / 256, 256, 0, stream>>>(emb, psrc, pdst,
                                                                nsrc, ndst, Z);
  dim3 g1((N_ROWS + 63) / 64, 64 / 32);
  gemm_wmma_f32<<<g1, 256, 0, stream>>>(Z, pW1, pb1, Z1, N_ROWS, 64, 128, 0.2f, 1);
  dim3 g2((N_ROWS + 63) / 64, 32 / 32);
  gemm_wmma_f32<<<g2, 256, 0, stream>>>(Z1, pW2, pb2, Z2, N_ROWS, 32, 64, 0.2f, 1);
  final_dot_kernel<<<(N_ROWS + 255) / 256, 256, 0, stream>>>(Z2, pW3, pb3, out);
}